// RefineDetFocalLoss_26920855011919
// MI455X (gfx1250) — compile-verified
//
#include <hip/hip_runtime.h>
#include <hip/hip_bf16.h>
#include <math.h>

#define B_ 32
#define P_ 32768
#define C_ 81
#define G_ 50
#define THRESH_ 0.5f
#define VAR0_ 0.1f
#define VAR1_ 0.2f
#define ALPHA_ 0.25f

typedef float v2f __attribute__((ext_vector_type(2)));
typedef float v8f __attribute__((ext_vector_type(8)));

// Wave32-wide fp32 sum using V_WMMA_F32_16X16X4_F32.
// A fragment: 16x4 fp32 = 2 VGPRs/lane. lane L<16 holds A[L][0],A[L][1];
// lane L>=16 holds A[L-16][2],A[L-16][3]. We put x in slot0, 0 in slot1,
// B = all ones -> D[M][N] = x_M + x_{M+16} broadcast over N.
// Lane sums its 8 D regs (M=0..7 or 8..15 halves), then xor-16 shuffle adds
// the halves: every lane ends with the exact fp32 sum of all 32 lanes.
__device__ __forceinline__ float wave_sum_wmma(float x) {
  v2f a = {x, 0.0f};
  v2f b = {1.0f, 1.0f};
  v8f c = {};
  c = __builtin_amdgcn_wmma_f32_16x16x4_f32(
      /*neg_a=*/false, a, /*neg_b=*/false, b,
      /*c_mod=*/(short)0, c, /*reuse_a=*/false, /*reuse_b=*/false);
  float s = c[0] + c[1] + c[2] + c[3] + c[4] + c[5] + c[6] + c[7];
  s += __shfl_xor(s, 16, 32);
  return s;
}

__device__ __forceinline__ float iou_pt(float ax0, float ay0, float ax1, float ay1,
                                        float bx0, float by0, float bx1, float by1) {
  float lx = fmaxf(ax0, bx0), ly = fmaxf(ay0, by0);
  float rx = fminf(ax1, bx1), ry = fminf(ay1, by1);
  float w = fmaxf(rx - lx, 0.0f), h = fmaxf(ry - ly, 0.0f);
  float inter = w * h;
  float aa = (ax1 - ax0) * (ay1 - ay0);
  float ab = (bx1 - bx0) * (by1 - by0);
  return inter / (aa + ab - inter);
}

__global__ void k_zero(float* accum) {
  if (threadIdx.x < 8) accum[threadIdx.x] = 0.0f;
}

// One block per (b,g): argmax over all P priors of IoU(truth, prior_pt).
// First-occurrence tie-break (match jnp.argmax).
__global__ void k_best_prior(const float* __restrict__ priors,
                             const float* __restrict__ targets,
                             int* __restrict__ bpi) {
  const int bg = blockIdx.x;  // b*G_ + g
  const float* t = targets + bg * 5;
  const float tx0 = t[0], ty0 = t[1], tx1 = t[2], ty1 = t[3];
  float bov = -1.0f;
  int bidx = 0x7fffffff;
  for (int p = threadIdx.x; p < P_; p += blockDim.x) {
    const float* pr = priors + p * 4;
    float cx = pr[0], cy = pr[1], w = pr[2], h = pr[3];
    float ov = iou_pt(tx0, ty0, tx1, ty1,
                      cx - 0.5f * w, cy - 0.5f * h, cx + 0.5f * w, cy + 0.5f * h);
    if (ov > bov) { bov = ov; bidx = p; }  // strict >: keeps smallest p
  }
  __shared__ float sov[256];
  __shared__ int sidx[256];
  const int tid = threadIdx.x;
  sov[tid] = bov; sidx[tid] = bidx;
  __syncthreads();
  for (int s = 128; s > 0; s >>= 1) {
    if (tid < s) {
      float o2 = sov[tid + s]; int i2 = sidx[tid + s];
      if (o2 > sov[tid] || (o2 == sov[tid] && i2 < sidx[tid])) {
        sov[tid] = o2; sidx[tid] = i2;
      }
    }
    __syncthreads();
  }
  if (tid == 0) bpi[bg] = sidx[0];
}

// One thread per (b,p): best truth over g, last-wins override scan,
// encode + smooth-L1, conf_t out. WMMA wave reduction for loss_l and N.
__global__ void k_match_locloss(const float* __restrict__ arm_loc,
                                const float* __restrict__ priors,
                                const float* __restrict__ targets,
                                const int* __restrict__ bpi,
                                int* __restrict__ conf_t,
                                float* __restrict__ accum) {
  const int b = blockIdx.x >> 7;         // / (P_/256) == 128
  const int chunk = blockIdx.x & 127;
  const int p = chunk * 256 + threadIdx.x;

  __shared__ float tr[G_ * 5];
  __shared__ int sbpi[G_];
  if (threadIdx.x < G_) {
    const float* t = targets + (b * G_ + threadIdx.x) * 5;
#pragma unroll
    for (int k = 0; k < 5; ++k) tr[threadIdx.x * 5 + k] = t[k];
    sbpi[threadIdx.x] = bpi[b * G_ + threadIdx.x];
  }
  __syncthreads();

  const float* pr = priors + p * 4;
  const float cx = pr[0], cy = pr[1], pw = pr[2], ph = pr[3];
  const float px0 = cx - 0.5f * pw, py0 = cy - 0.5f * ph;
  const float px1 = cx + 0.5f * pw, py1 = cy + 0.5f * ph;

  float bov = -1.0f;
  int bidx = 0;
  for (int g = 0; g < G_; ++g) {
    float ov = iou_pt(tr[g * 5], tr[g * 5 + 1], tr[g * 5 + 2], tr[g * 5 + 3],
                      px0, py0, px1, py1);
    if (ov > bov) { bov = ov; bidx = g; }  // first occurrence
  }
  // Sequential last-wins override: mimics .at[best_prior_idx].set(arange(G))
  for (int g = 0; g < G_; ++g) {
    if (sbpi[g] == p) { bidx = g; bov = 2.0f; }
  }

  int conf = (int)tr[bidx * 5 + 4] + 1;
  if (bov < THRESH_) conf = 0;
  conf_t[b * P_ + p] = conf;

  float lsum = 0.0f, np = 0.0f;
  if (conf > 0) {
    float mx0 = tr[bidx * 5], my0 = tr[bidx * 5 + 1];
    float mx1 = tr[bidx * 5 + 2], my1 = tr[bidx * 5 + 3];
    float lt[4];
    lt[0] = ((mx0 + mx1) * 0.5f - cx) / (VAR0_ * pw);
    lt[1] = ((my0 + my1) * 0.5f - cy) / (VAR0_ * ph);
    lt[2] = __logf((mx1 - mx0) / pw) / VAR1_;
    lt[3] = __logf((my1 - my0) / ph) / VAR1_;
    const float* al = arm_loc + (size_t)(b * P_ + p) * 4;
#pragma unroll
    for (int k = 0; k < 4; ++k) {
      float d = fabsf(al[k] - lt[k]);
      lsum += (d < 1.0f) ? 0.5f * d * d : d - 0.5f;
    }
    np = 1.0f;
  }
  float wl = wave_sum_wmma(lsum);
  float wn = wave_sum_wmma(np);
  if ((threadIdx.x & 31) == 0) {
    atomicAdd(&accum[0], wl);
    atomicAdd(&accum[1], wn);
  }
}

// One wave per 81-class row; fully coalesced lane loads; shuffle softmax;
// matched-class logit via __shfl (zero extra traffic). Per-row contribution
// rotated across lanes so the terminal WMMA reduction is a real 32-way sum.
__global__ void k_focal(const float* __restrict__ conf_data,
                        const int* __restrict__ conf_t,
                        float* __restrict__ accum) {
  const int lane = threadIdx.x & 31;
  const int wid = (blockIdx.x * blockDim.x + threadIdx.x) >> 5;
  const int nw = (gridDim.x * blockDim.x) >> 5;
  float acc = 0.0f;
  for (int row = wid; row < B_ * P_; row += nw) {
    const float* x = conf_data + (size_t)row * C_;
    if (row + nw < B_ * P_)
      __builtin_prefetch(conf_data + (size_t)(row + nw) * C_, 0, 0);
    const bool has2 = lane < (C_ - 64);  // lanes 0..16
    float x0 = x[lane];
    float x1 = x[lane + 32];
    float x2 = has2 ? x[lane + 64] : -3.4e38f;
    float m = fmaxf(fmaxf(x0, x1), x2);
#pragma unroll
    for (int off = 16; off > 0; off >>= 1) m = fmaxf(m, __shfl_xor(m, off, 32));
    float s = __expf(x0 - m) + __expf(x1 - m) + (has2 ? __expf(x2 - m) : 0.0f);
#pragma unroll
    for (int off = 16; off > 0; off >>= 1) s += __shfl_xor(s, off, 32);
    int t = conf_t[row];
    float xt;
    if (t < 32)      xt = __shfl(x0, t, 32);
    else if (t < 64) xt = __shfl(x1, t - 32, 32);
    else             xt = __shfl(x2, t - 64, 32);
    float logp = (xt - m) - __logf(s);
    float pt = __expf(xt - m) / s;
    float om = 1.0f - pt;
    float contrib = logp * (-ALPHA_) * om * om;
    if (lane == (row & 31)) acc += contrib;
  }
  float tot = wave_sum_wmma(acc);
  if (lane == 0) atomicAdd(&accum[2], tot);
}

__global__ void k_final(const float* __restrict__ accum, float* __restrict__ out) {
  if (threadIdx.x == 0) {
    out[0] = accum[0] / accum[1];          // loss_l / N
    out[1] = accum[2] / (float)P_;         // loss_c
  }
}

extern "C" void kernel_launch(void* const* d_in, const int* in_sizes, int n_in,
                              void* d_out, int out_size, void* d_ws, size_t ws_size,
                              hipStream_t stream) {
  const float* arm_loc  = (const float*)d_in[0];
  const float* arm_conf = (const float*)d_in[1];
  // d_in[2], d_in[3] = odm_loc/odm_conf: unused by the reference loss path
  const float* priors   = (const float*)d_in[4];
  const float* targets  = (const float*)d_in[5];
  float* out = (float*)d_out;

  char* ws = (char*)d_ws;
  float* accum  = (float*)ws;                         // 32 B
  int*   bpi    = (int*)(ws + 64);                    // B*G ints = 6400 B
  int*   conf_t = (int*)(ws + 6464);                  // B*P ints = 4 MB

  k_zero<<<1, 32, 0, stream>>>(accum);
  k_best_prior<<<B_ * G_, 256, 0, stream>>>(priors, targets, bpi);
  k_match_locloss<<<B_ * (P_ / 256), 256, 0, stream>>>(arm_loc, priors, targets,
                                                       bpi, conf_t, accum);
  k_focal<<<4096, 256, 0, stream>>>(arm_conf, conf_t, accum);
  k_final<<<1, 32, 0, stream>>>(accum, out);
}